// MarginRankingLossPairs_54408645706296
// MI455X (gfx1250) — compile-verified
//
#include <hip/hip_runtime.h>

typedef __attribute__((ext_vector_type(2))) float v2f;
typedef __attribute__((ext_vector_type(8))) float v8f;

#define MARGIN_F 0.1f
#define B_ 32
#define S_ 2048
#define TILES_PER_ROW (S_ / 16)          // 128
#define NSTRIPS (B_ * TILES_PER_ROW)     // 4096 waves total
#define WAVES_PER_BLOCK 8                // 256 threads, wave32

// One wave handles a 16-row strip [i0, i0+16) of batch row b, sweeping all
// j-tiles with j0 >= i0. Pairwise difference tiles (scores and targets) are
// produced on the matrix pipe via rank-2 V_WMMA_F32_16X16X4_F32; VALU only
// does sign/hinge/accumulate on the 8 C-layout VGPRs.
__global__ __launch_bounds__(256) void mrl_pairs_wmma_kernel(
    const float* __restrict__ scores,
    const int*   __restrict__ targets,
    float*       __restrict__ partial) {
  const int lane      = threadIdx.x & 31;
  const int waveInBlk = threadIdx.x >> 5;
  const int gw = blockIdx.x * WAVES_PER_BLOCK + waveInBlk;   // 0..NSTRIPS-1
  const int b  = gw >> 7;                                    // / TILES_PER_ROW
  const int ti = gw & (TILES_PER_ROW - 1);
  const int i0 = ti << 4;

  const float* __restrict__ srow = scores  + b * S_;
  const int*   __restrict__ trow = targets + b * S_;

  const int  lj = lane & 15;      // tile column N owned by this lane (C/D layout)
  const bool hi = lane >= 16;     // lanes 16-31 hold rows M=8..15 of C/D

  // A matrices (hoisted, loop-invariant). 16x4 f32 A layout:
  //   VGPR0: lanes0-15 = K0 col, lanes16-31 = K2 col (zeroed)
  //   VGPR1: lanes0-15 = K1 col, lanes16-31 = K3 col (zeroed)
  // A[:,0] = value_i, A[:,1] = 1  ->  D = value_i*B[0][:] + B[1][:]
  const float si = srow[i0 + lj];
  const float tiv = (float)trow[i0 + lj];   // exact: |t| < 2048
  v2f As, At;
  As.x = hi ? 0.0f : si;   As.y = hi ? 0.0f : 1.0f;
  At.x = hi ? 0.0f : tiv;  At.y = hi ? 0.0f : 1.0f;

  v8f acc = {};

  for (int j0 = i0; j0 < S_; j0 += 16) {
    // gfx1250 prefetch of the next j-tile line (global_prefetch_b8)
    if (j0 + 16 < S_) {
      __builtin_prefetch(&srow[j0 + 16 + lj], 0, 3);
      __builtin_prefetch(&trow[j0 + 16 + lj], 0, 3);
    }

    const float sj = srow[j0 + lj];
    const float tj = (float)trow[j0 + lj];
    // B matrices: B[0][j] = 1, B[1][j] = -value_j, K2/K3 rows zero.
    v2f Bs, Bt;
    Bs.x = hi ? 0.0f : 1.0f;  Bs.y = hi ? 0.0f : -sj;
    Bt.x = hi ? 0.0f : 1.0f;  Bt.y = hi ? 0.0f : -tj;

    v8f zero = {};
    // sdiff[i][j] = s_i - s_j ; tdiff[i][j] = t_i - t_j   (matrix pipe)
    v8f sd = __builtin_amdgcn_wmma_f32_16x16x4_f32(
        false, As, false, Bs, (short)0, zero, false, false);
    v8f td = __builtin_amdgcn_wmma_f32_16x16x4_f32(
        false, At, false, Bt, (short)0, zero, false, false);

    if (j0 == i0) {
      // Diagonal tile: keep only strictly-upper elements (i < j).
#pragma unroll
      for (int k = 0; k < 8; ++k) {
        const int il = k + (hi ? 8 : 0);          // local row M
        float y = (td[k] > 0.0f) ? 1.0f : ((td[k] < 0.0f) ? -1.0f : 0.0f);
        float v = fmaxf(0.0f, MARGIN_F - y * sd[k]);
        acc[k] += (il < lj) ? v : 0.0f;
      }
    } else {
#pragma unroll
      for (int k = 0; k < 8; ++k) {
        float y = (td[k] > 0.0f) ? 1.0f : ((td[k] < 0.0f) ? -1.0f : 0.0f);
        acc[k] += fmaxf(0.0f, MARGIN_F - y * sd[k]);
      }
    }
  }

  // Wave-local reduction: 8 accumulators, then 32-lane xor-shuffle tree.
  float sum = acc[0] + acc[1] + acc[2] + acc[3] +
              acc[4] + acc[5] + acc[6] + acc[7];
#pragma unroll
  for (int off = 16; off > 0; off >>= 1)
    sum += __shfl_xor(sum, off, 32);

  if (lane == 0) partial[gw] = sum;
}

// Deterministic final reduction: single block, fixed-order strided sums +
// LDS tree; writes the scaled scalar. No float atomics anywhere.
__global__ __launch_bounds__(256) void mrl_reduce_kernel(
    const float* __restrict__ partial, float* __restrict__ out) {
  __shared__ float smem[256];
  float v = 0.0f;
  for (int i = threadIdx.x; i < NSTRIPS; i += 256) v += partial[i];
  smem[threadIdx.x] = v;
  __syncthreads();
  for (int step = 128; step > 0; step >>= 1) {
    if ((int)threadIdx.x < step) smem[threadIdx.x] += smem[threadIdx.x + step];
    __syncthreads();
  }
  if (threadIdx.x == 0) {
    // scale = 1 / (B * S*(S-1)/2) = 1 / (32 * 2096128)
    const float scale = 1.0f / 67076096.0f;
    out[0] = smem[0] * scale;
  }
}

extern "C" void kernel_launch(void* const* d_in, const int* in_sizes, int n_in,
                              void* d_out, int out_size, void* d_ws, size_t ws_size,
                              hipStream_t stream) {
  const float* scores  = (const float*)d_in[0];
  const int*   targets = (const int*)d_in[1];
  float*       out     = (float*)d_out;
  float*       partial = (float*)d_ws;    // NSTRIPS floats = 16 KB scratch

  mrl_pairs_wmma_kernel<<<NSTRIPS / WAVES_PER_BLOCK, 256, 0, stream>>>(
      scores, targets, partial);
  mrl_reduce_kernel<<<1, 256, 0, stream>>>(partial, out);
}